// QFNN_81604378624423
// MI455X (gfx1250) — compile-verified
//
#include <hip/hip_runtime.h>

// QFNN-6 closed form, per subsystem s (x0=ty_a, x1=tz_a, x2=ty_b, x3=tz_b):
//   Z_even = cos(x0)*cos(x2)
//   Z_odd  = cos(p1+p2)*cos(x2) - sin(p1+p2)*sin(x0)*sin(x2)*sin(x3)*cos(x1+p0)
// with cos(x1+p0) expanded as cos(x1)cos(p0) - sin(x1)sin(p0) so that ALL
// per-thread trig arguments lie in [0,1) -> plain Taylor polys, no reduction.

// Taylor sin/cos, accurate to ~3e-8 absolute on |x| <= 1 (trunc: x^11/11!, x^12/12!).
__device__ __forceinline__ void sincos_poly01(float x, float* s, float* c) {
    const float z = x * x;
    float ps = fmaf(z, fmaf(z, fmaf(z, fmaf(z, 2.7557319224e-6f,
                                            -1.9841269841e-4f),
                                     8.3333333333e-3f),
                             -1.6666666667e-1f), 1.0f);
    *s = x * ps;
    *c = fmaf(z, fmaf(z, fmaf(z, fmaf(z, fmaf(z, -2.7557319224e-7f,
                                              2.4801587302e-5f),
                                       -1.3888888889e-3f),
                               4.1666666667e-2f),
                       -0.5f), 1.0f);
}

__device__ __forceinline__ float sin_poly01(float x) {
    const float z = x * x;
    return x * fmaf(z, fmaf(z, fmaf(z, fmaf(z, 2.7557319224e-6f,
                                            -1.9841269841e-4f),
                                     8.3333333333e-3f),
                             -1.6666666667e-1f), 1.0f);
}

// Cody-Waite reduced sincos for the (small, wave-uniform) weight angles.
// Good for |x| up to a few hundred; weights are N(0,1)-scale.
__device__ __forceinline__ void sincos_cw(float x, float* s, float* c) {
    const float k = rintf(x * 0.6366197723675814f);        // x * 2/pi
    float r = fmaf(-k, 1.57079637050628662e+0f, x);        // pi/2 hi (0x3FC90FDB)
    r = fmaf(k, 4.37113882867379e-8f, r);                  // pi/2 = hi - lo
    float ss, cc;
    sincos_poly01(r, &ss, &cc);                            // |r| <= pi/4 < 1
    const int q = ((int)k) & 3;
    float sr = (q & 1) ? cc : ss;
    float cr = (q & 1) ? ss : cc;
    if (q == 1 || q == 2) cr = -cr;
    if (q >= 2) sr = -sr;
    *s = sr;
    *c = cr;
}

__global__ __launch_bounds__(256) void qfnn6_kernel(const float* __restrict__ x,
                                                    const float* __restrict__ w,
                                                    float* __restrict__ out,
                                                    int B) {
    // Wave-uniform weights -> scalar loads; trig computed once per wave.
    float sp0[3], cp0[3], cP[3], sP[3];
#pragma unroll
    for (int s = 0; s < 3; ++s) {
        sincos_cw(w[3 * s + 0], &sp0[s], &cp0[s]);
        sincos_cw(w[3 * s + 1] + w[3 * s + 2], &sP[s], &cP[s]);
    }

    int i = blockIdx.x * blockDim.x + threadIdx.x;
    const int stride = gridDim.x * blockDim.x;

    for (; i < B; i += stride) {
        // Row is 48 bytes, 16B-aligned -> three global_load_b128.
        const float4* __restrict__ row = (const float4*)(x + (size_t)i * 12);
        float4 v0 = row[0];
        float4 v1 = row[1];
        float4 v2 = row[2];

        // gfx1250 prefetch path for a (potential) next grid-stride tile.
        if (i + stride < B) {
            __builtin_prefetch(x + (size_t)(i + stride) * 12, 0, 0);
        }

        const float r[12] = {v0.x, v0.y, v0.z, v0.w,
                             v1.x, v1.y, v1.z, v1.w,
                             v2.x, v2.y, v2.z, v2.w};
        float o[6];
#pragma unroll
        for (int s = 0; s < 3; ++s) {
            const float x0 = r[4 * s + 0];
            const float x1 = r[4 * s + 1];
            const float x2 = r[4 * s + 2];
            const float x3 = r[4 * s + 3];

            float s0, c0, s1, c1, s2, c2;
            sincos_poly01(x0, &s0, &c0);
            sincos_poly01(x1, &s1, &c1);
            sincos_poly01(x2, &s2, &c2);
            const float s3 = sin_poly01(x3);

            // cos(x1 + p0) via angle addition with uniform sin/cos(p0)
            const float cA = fmaf(c1, cp0[s], -(s1 * sp0[s]));

            o[2 * s + 0] = c0 * c2;
            o[2 * s + 1] = fmaf(cP[s], c2, -(sP[s] * s0 * s2 * s3 * cA));
        }

        // 24-byte rows, 8B-aligned -> three global_store_b64.
        float2* __restrict__ orow = (float2*)(out + (size_t)i * 6);
        orow[0] = make_float2(o[0], o[1]);
        orow[1] = make_float2(o[2], o[3]);
        orow[2] = make_float2(o[4], o[5]);
    }
}

extern "C" void kernel_launch(void* const* d_in, const int* in_sizes, int n_in,
                              void* d_out, int out_size, void* d_ws, size_t ws_size,
                              hipStream_t stream) {
    const float* x = (const float*)d_in[0];   // [B, 12] fp32
    const float* w = (const float*)d_in[1];   // [9] fp32
    float* out = (float*)d_out;               // [B, 6] fp32

    const int B = in_sizes[0] / 12;
    const int block = 256;                    // 8 wave32s
    const int grid = (B + block - 1) / block; // 1024 blocks for B = 262144

    qfnn6_kernel<<<grid, block, 0, stream>>>(x, w, out, B);
}